// homoGraph_54812372631568
// MI455X (gfx1250) — compile-verified
//
#include <hip/hip_runtime.h>

#define D 128
#define WL_STRIDE 136   // LDS row stride for W tile (bank-conflict shift of 16 between lane halves)
#define XL_STRIDE 132   // LDS row stride for x tile (float4-aligned, conflict-free)

typedef float v2f __attribute__((ext_vector_type(2)));
typedef float v8f __attribute__((ext_vector_type(8)));

// ------------------------------------------------------------------
// Degree / symmetric norm:  deg[v] = 1 (self loop) + #edges with col==v
// then dinv[v] = deg > 0 ? deg^-1/2 : 0   (in place)
// ------------------------------------------------------------------
__global__ void deg_init_kernel(float* __restrict__ deg, int n) {
  int v = blockIdx.x * blockDim.x + threadIdx.x;
  if (v < n) deg[v] = 1.0f;               // self-loop weight
}

__global__ void deg_edge_kernel(const long long* __restrict__ col,
                                float* __restrict__ deg, int e) {
  int i = blockIdx.x * blockDim.x + threadIdx.x;
  if (i < e) atomicAdd(&deg[col[i]], 1.0f);
}

__global__ void dinv_kernel(float* __restrict__ deg, int n) {
  int v = blockIdx.x * blockDim.x + threadIdx.x;
  if (v < n) {
    float d = deg[v];
    deg[v] = (d > 0.0f) ? (1.0f / sqrtf(d)) : 0.0f;
  }
}

// ------------------------------------------------------------------
// H = X @ W   (M x 128) @ (128 x 128), fp32, via V_WMMA_F32_16X16X4_F32.
// Block = 256 threads = 8 waves; block owns a 16-row stripe of X,
// wave w owns 16-column tile w. W and the X stripe are staged in LDS.
// ------------------------------------------------------------------
__global__ __launch_bounds__(256)
void gemm_wmma_kernel(const float* __restrict__ X, const float* __restrict__ W,
                      float* __restrict__ H, int M) {
  __shared__ float Wl[128 * WL_STRIDE];
  __shared__ float Xl[16 * XL_STRIDE];

  const int t     = threadIdx.x;
  const int mbase = blockIdx.x * 16;

  // Stage W (128x128) into LDS: 4096 float4, 16 per thread.
  for (int i = t; i < 128 * 32; i += 256) {
    int r  = i >> 5;
    int c4 = i & 31;
    float4 w = ((const float4*)(W + (size_t)r * D))[c4];
    float* dst = &Wl[r * WL_STRIDE + c4 * 4];
    dst[0] = w.x; dst[1] = w.y; dst[2] = w.z; dst[3] = w.w;
  }
  // Stage 16 x 128 X stripe: 512 float4, 2 per thread (clamp rows for safety).
  for (int i = t; i < 16 * 32; i += 256) {
    int r  = i >> 5;
    int c4 = i & 31;
    int rr = mbase + r; if (rr >= M) rr = M - 1;
    float4 xv = ((const float4*)(X + (size_t)rr * D))[c4];
    *(float4*)&Xl[r * XL_STRIDE + c4 * 4] = xv;
  }
  __syncthreads();

  const int wave  = t >> 5;          // 0..7 -> output column tile
  const int lane  = t & 31;
  const int nbase = wave * 16;
  const int lrow  = lane & 15;       // M (for A) / N (for B,C) within tile
  const int khalf = (lane >> 4) * 2; // lanes 16..31 hold K+2,K+3

  v8f c = {};                        // zero accumulator
#pragma unroll
  for (int kk = 0; kk < 32; ++kk) {
    const int kb = kk * 4 + khalf;
    v2f a, b;
    // A 16x4 f32: lane L holds A[L&15][kb], A[L&15][kb+1]
    a[0] = Xl[lrow * XL_STRIDE + kb];
    a[1] = Xl[lrow * XL_STRIDE + kb + 1];
    // B 4x16 f32: lane L holds B[kb][nbase + (L&15)], B[kb+1][...]
    b[0] = Wl[kb * WL_STRIDE + nbase + lrow];
    b[1] = Wl[(kb + 1) * WL_STRIDE + nbase + lrow];
    c = __builtin_amdgcn_wmma_f32_16x16x4_f32(false, a, false, b,
                                              (short)0, c, false, false);
  }

  // C/D 16x16 f32: lane L, VGPR r -> (M = (L>>4)*8 + r, N = L&15)
  const int col   = nbase + lrow;
  const int rbase = mbase + (lane >> 4) * 8;
#pragma unroll
  for (int r = 0; r < 8; ++r) {
    int rr = rbase + r;
    if (rr < M) H[(size_t)rr * D + col] = c[r];
  }
}

// ------------------------------------------------------------------
// agg[v] = dinv[v]^2 * h[v]   (self-loop term, also zero-fills agg)
// ------------------------------------------------------------------
__global__ void agg_init_kernel(const float* __restrict__ h,
                                const float* __restrict__ dinv,
                                float* __restrict__ agg, int n) {
  int i = blockIdx.x * blockDim.x + threadIdx.x;   // float4 index
  if (i >= n * (D / 4)) return;
  int v = i >> 5;
  float dv = dinv[v];
  float s = dv * dv;
  float4 hv = ((const float4*)h)[i];
  float4 o = {s * hv.x, s * hv.y, s * hv.z, s * hv.w};
  ((float4*)agg)[i] = o;
}

// ------------------------------------------------------------------
// One wave per edge: lane k moves float4 k of the 128-float payload.
// agg[col] += dinv[row]*dinv[col] * h[row]   (fp32 global atomics, L2-resident)
// ------------------------------------------------------------------
__global__ __launch_bounds__(256)
void edge_agg_kernel(const float* __restrict__ h, const float* __restrict__ dinv,
                     const long long* __restrict__ row,
                     const long long* __restrict__ col,
                     float* __restrict__ agg, int e) {
  int warp = (int)((blockIdx.x * (unsigned)blockDim.x + threadIdx.x) >> 5);
  int lane = threadIdx.x & 31;
  if (warp >= e) return;
  long long r = row[warp];
  long long c = col[warp];
  float nrm = dinv[r] * dinv[c];
  float4 v = ((const float4*)(h + (size_t)r * D))[lane];
  float* dst = agg + (size_t)c * D + lane * 4;
  atomicAdd(dst + 0, nrm * v.x);
  atomicAdd(dst + 1, nrm * v.y);
  atomicAdd(dst + 2, nrm * v.z);
  atomicAdd(dst + 3, nrm * v.w);
}

// ------------------------------------------------------------------
// agg = relu(agg + bias)  (in place)
// ------------------------------------------------------------------
__global__ void finalize_relu_kernel(float* __restrict__ agg,
                                     const float* __restrict__ bias, int n) {
  int i = blockIdx.x * blockDim.x + threadIdx.x;   // float4 index
  if (i >= n * (D / 4)) return;
  int d4 = (i & 31) * 4;
  float4 a = ((float4*)agg)[i];
  float4 b = *(const float4*)(bias + d4);
  a.x = fmaxf(a.x + b.x, 0.0f);
  a.y = fmaxf(a.y + b.y, 0.0f);
  a.z = fmaxf(a.z + b.z, 0.0f);
  a.w = fmaxf(a.w + b.w, 0.0f);
  ((float4*)agg)[i] = a;
}

// ------------------------------------------------------------------
extern "C" void kernel_launch(void* const* d_in, const int* in_sizes, int n_in,
                              void* d_out, int out_size, void* d_ws, size_t ws_size,
                              hipStream_t stream) {
  const float*     x   = (const float*)d_in[0];
  const long long* ei  = (const long long*)d_in[1];   // int64 [2, E]
  const float*     W0  = (const float*)d_in[2];
  const float*     b0  = (const float*)d_in[3];
  const float*     W1  = (const float*)d_in[4];
  const float*     b1  = (const float*)d_in[5];
  const float*     W2  = (const float*)d_in[6];
  const float*     b2  = (const float*)d_in[7];

  const int N = in_sizes[0] / D;
  const int E = in_sizes[1] / 2;
  const long long* rowp = ei;
  const long long* colp = ei + E;

  // Workspace: dinv[N] | bufH[N*D] | bufX[N*D]
  char* ws = (char*)d_ws;
  size_t off = 0;
  float* dinv = (float*)(ws + off); off += (((size_t)N * 4) + 255) & ~(size_t)255;
  float* bufH = (float*)(ws + off); off += (size_t)N * D * 4;
  float* bufX = (float*)(ws + off);

  const dim3 blk(256);

  // Norm (computed once, shared across the 3 layers as in the reference).
  deg_init_kernel<<<(N + 255) / 256, blk, 0, stream>>>(dinv, N);
  deg_edge_kernel<<<(E + 255) / 256, blk, 0, stream>>>(colp, dinv, E);
  dinv_kernel   <<<(N + 255) / 256, blk, 0, stream>>>(dinv, N);

  const int mtiles     = (N + 15) / 16;
  const int vec4       = N * (D / 4);
  const int aggblocks  = (vec4 + 255) / 256;
  const int edgeblocks = (E + 7) / 8;          // 8 waves (edges) per block

  const float* Ws[3] = {W0, W1, W2};
  const float* bs[3] = {b0, b1, b2};
  const float* xin = x;
  for (int l = 0; l < 3; ++l) {
    // Final layer aggregates straight into d_out; earlier layers ping-pong bufX.
    float* aggbuf = (l == 2) ? (float*)d_out : bufX;
    gemm_wmma_kernel    <<<mtiles, blk, 0, stream>>>(xin, Ws[l], bufH, N);
    agg_init_kernel     <<<aggblocks, blk, 0, stream>>>(bufH, dinv, aggbuf, N);
    edge_agg_kernel     <<<edgeblocks, blk, 0, stream>>>(bufH, dinv, rowp, colp, aggbuf, E);
    finalize_relu_kernel<<<aggblocks, blk, 0, stream>>>(aggbuf, bs[l], N);
    xin = aggbuf;
  }
}